// PaDiM_20229295964482
// MI455X (gfx1250) — compile-verified
//
#include <hip/hip_runtime.h>

// PaDiM accumulation on MI455X (gfx1250, wave32):
//   cov[p] += X_p^T X_p  (X_p = emb[:, :, p], 32x200),  means[p] += sum_b X_p
// Bandwidth-bound (~1.08 GB mandatory HBM traffic, ~46us floor @ 23.3 TB/s);
// compute via V_WMMA_F32_16X16X4_F32 to match f32 reference exactly.

typedef float v2f __attribute__((ext_vector_type(2)));
typedef float v8f __attribute__((ext_vector_type(8)));

#define B_DIM 32
#define E_DIM 200
#define P_DIM 3136
#define E_PAD 208          // 13 * 16, zero-padded
#define NTOT  169          // 13*13 output tiles of 16x16

__global__ __launch_bounds__(256)
void padim_cov_wmma(const float* __restrict__ emb,      // [B, E, P]
                    const float* __restrict__ cov_in,   // [P, E, E]
                    const float* __restrict__ mean_in,  // [P, E]
                    float* __restrict__ cov_out,        // [P, E, E]
                    float* __restrict__ mean_out)       // [P, E]
{
    __shared__ float Xs[B_DIM][E_PAD];   // 32*208*4 = 26,624 B per workgroup
    const int p   = blockIdx.x;
    const int tid = threadIdx.x;

    // Stage X_p into LDS (strided gather over [b,e] at fixed p; lines are
    // shared with 15 neighboring patches via the 192MB L2 — emb fits in L2).
    for (int i = tid; i < B_DIM * E_PAD; i += 256) {
        const int b = i / E_PAD;
        const int e = i - b * E_PAD;
        float v = 0.0f;
        if (e < E_DIM)
            v = emb[(size_t)(b * E_DIM + e) * P_DIM + p];
        Xs[b][e] = v;
    }
    __syncthreads();

    const int lane = tid & 31;
    const int wave = tid >> 5;
    const int m    = lane & 15;   // A-row / B-col / C-col within tile
    const int kh   = lane >> 4;   // half-wave: K pair select, C row half

    const size_t covBase = (size_t)p * E_DIM * E_DIM;

    // Each of the 8 waves handles tiles t = wave, wave+8, ... (wave-uniform
    // control flow -> EXEC is all-ones at every WMMA).
    for (int t = wave; t < NTOT; t += 8) {
        const int ti = t / 13;
        const int tj = t - ti * 13;
        const int eA = ti * 16;   // C rows    (A M-dim)
        const int eB = tj * 16;   // C columns (B N-dim)

        // Accumulator tile = cov_in tile (non-temporal: streamed once).
        v8f c;
#pragma unroll
        for (int r = 0; r < 8; ++r) {
            const int row = eA + r + 8 * kh;
            const int col = eB + m;
            float cv = 0.0f;
            if (row < E_DIM && col < E_DIM)
                cv = __builtin_nontemporal_load(
                        &cov_in[covBase + (size_t)row * E_DIM + col]);
            c[r] = cv;
        }

        // Rank-32 update: 8 chained V_WMMA_F32_16X16X4_F32.
        // A (16x4): lane m holds rows, VGPR0/1 = K = 2*kh, 2*kh+1 of [bk..bk+3].
        // B (4x16): mirrored, lane m is the N index.
#pragma unroll
        for (int bk = 0; bk < B_DIM; bk += 4) {
            v2f av, bv;
            av.x = Xs[bk + 2 * kh + 0][eA + m];
            av.y = Xs[bk + 2 * kh + 1][eA + m];
            bv.x = Xs[bk + 2 * kh + 0][eB + m];
            bv.y = Xs[bk + 2 * kh + 1][eB + m];
            c = __builtin_amdgcn_wmma_f32_16x16x4_f32(
                    false, av, false, bv, (short)0, c, false, false);
        }

        // Store tile (non-temporal), predicated on the 200-edge.
#pragma unroll
        for (int r = 0; r < 8; ++r) {
            const int row = eA + r + 8 * kh;
            const int col = eB + m;
            if (row < E_DIM && col < E_DIM)
                __builtin_nontemporal_store(
                    c[r], &cov_out[covBase + (size_t)row * E_DIM + col]);
        }
    }

    // means[p] epilogue: 200 threads reduce the 32 LDS rows.
    if (tid < E_DIM) {
        float s = 0.0f;
#pragma unroll
        for (int b = 0; b < B_DIM; ++b)
            s += Xs[b][tid];
        mean_out[(size_t)p * E_DIM + tid] =
            mean_in[(size_t)p * E_DIM + tid] + s;
    }
}

extern "C" void kernel_launch(void* const* d_in, const int* in_sizes, int n_in,
                              void* d_out, int out_size, void* d_ws, size_t ws_size,
                              hipStream_t stream) {
    const float* emb     = (const float*)d_in[0];  // embeddings  [32,200,3136]
    const float* cov_in  = (const float*)d_in[1];  // covariances [3136,200,200]
    const float* mean_in = (const float*)d_in[2];  // means       [3136,200]

    float* cov_out  = (float*)d_out;
    float* mean_out = cov_out + (size_t)P_DIM * E_DIM * E_DIM;

    padim_cov_wmma<<<P_DIM, 256, 0, stream>>>(emb, cov_in, mean_in,
                                              cov_out, mean_out);
    (void)in_sizes; (void)n_in; (void)out_size; (void)d_ws; (void)ws_size;
}